// HeteroGnnEncoder_9706626089349
// MI455X (gfx1250) — compile-verified
//
#include <hip/hip_runtime.h>
#include <hip/hip_bf16.h>
#include <math.h>

typedef float v2f __attribute__((ext_vector_type(2)));
typedef float v8f __attribute__((ext_vector_type(8)));

#define DD 128

// ---------------------------------------------------------------------------
// C[M x 128] = A[M x 128] * W^T  (W: [128 x 128], row-major "out_d x in_d")
// optional bias[n] and residual res[m*128+n].  One wave per 16x64 tile.
// Uses V_WMMA_F32_16X16X4_F32 (f32 matrix core, full precision).
// A-frag: lane<16 -> K={0,1}, lane>=16 -> K={2,3} (ISA 7.12.2 32-bit A 16x4)
// B-frag: B[k][n] = W[n][k]; lanes hold col N=lane%16, K pair by lane half.
// ---------------------------------------------------------------------------
__global__ __launch_bounds__(32)
void gemm_nt_wmma(const float* __restrict__ A, const float* __restrict__ W,
                  float* __restrict__ C, int M,
                  const float* __restrict__ bias, const float* __restrict__ res)
{
    const int lane = threadIdx.x & 31;
    const int half = lane >> 4;      // 0: lanes 0-15, 1: lanes 16-31
    const int lrow = lane & 15;
    const int m0 = blockIdx.x << 4;
    const int n0 = blockIdx.y << 6;

    const float* Arow = A + (size_t)(m0 + lrow) * DD;
    const float* W0   = W + (size_t)(n0 + lrow) * DD;
    const float* W1   = W0 + 16 * DD;
    const float* W2   = W0 + 32 * DD;
    const float* W3   = W0 + 48 * DD;

    v8f acc0 = {}; v8f acc1 = {}; v8f acc2 = {}; v8f acc3 = {};

#pragma unroll 4
    for (int k0 = 0; k0 < DD; k0 += 4) {
        const int kk = k0 + (half << 1);
        v2f a  = *(const v2f*)(Arow + kk);
        v2f b0 = *(const v2f*)(W0 + kk);
        v2f b1 = *(const v2f*)(W1 + kk);
        v2f b2 = *(const v2f*)(W2 + kk);
        v2f b3 = *(const v2f*)(W3 + kk);
        acc0 = __builtin_amdgcn_wmma_f32_16x16x4_f32(false, a, false, b0, (short)0, acc0, false, false);
        acc1 = __builtin_amdgcn_wmma_f32_16x16x4_f32(false, a, false, b1, (short)0, acc1, false, false);
        acc2 = __builtin_amdgcn_wmma_f32_16x16x4_f32(false, a, false, b2, (short)0, acc2, false, false);
        acc3 = __builtin_amdgcn_wmma_f32_16x16x4_f32(false, a, false, b3, (short)0, acc3, false, false);
    }

#pragma unroll
    for (int r = 0; r < 8; ++r) {
        const int row = m0 + r + (half << 3);
        const size_t base = (size_t)row * DD + n0 + lrow;
        float v0 = acc0[r], v1 = acc1[r], v2 = acc2[r], v3 = acc3[r];
        if (bias) {
            v0 += bias[n0 + lrow];      v1 += bias[n0 + 16 + lrow];
            v2 += bias[n0 + 32 + lrow]; v3 += bias[n0 + 48 + lrow];
        }
        if (res) {
            v0 += res[base]; v1 += res[base + 16]; v2 += res[base + 32]; v3 += res[base + 48];
        }
        C[base] = v0; C[base + 16] = v1; C[base + 32] = v2; C[base + 48] = v3;
    }
}

// out[n] = dot(H[n, :128], att).  One wave per node, 4 nodes per block.
__global__ __launch_bounds__(128)
void rowdot(const float* __restrict__ H, const float* __restrict__ att,
            float* __restrict__ out, int N)
{
    const int lane = threadIdx.x & 31;
    const int wid  = threadIdx.x >> 5;
    const int n = blockIdx.x * 4 + wid;
    if (n >= N) return;
    const float4 h = *(const float4*)(H + (size_t)n * DD + lane * 4);
    const float4 a = *(const float4*)(att + lane * 4);
    float s = h.x * a.x + h.y * a.y + h.z * a.z + h.w * a.w;
    for (int m = 16; m; m >>= 1) s += __shfl_xor(s, m, 32);
    if (lane == 0) out[n] = s;
}

// c = dot(att_e, We[:,0])  (single scalar per relation)
__global__ __launch_bounds__(32)
void dotc_kernel(const float* __restrict__ att_e, const float* __restrict__ We,
                 float* __restrict__ out)
{
    const int lane = threadIdx.x;
    const float4 a = *(const float4*)(att_e + lane * 4);
    const float4 w = *(const float4*)(We + lane * 4);
    float s = a.x * w.x + a.y * w.y + a.z * w.z + a.w * w.w;
    for (int m = 16; m; m >>= 1) s += __shfl_xor(s, m, 32);
    if (lane == 0) out[0] = s;
}

__device__ __forceinline__ float lrelu2(float a) { return a > 0.f ? a : 0.2f * a; }

// Full-bipartite relation (vp_im / pv_im). One block (128 thr) per dst node.
// edge id = g*NSg*NDg + i*eS + j*eD;  src node = g*NSg + i, i = 0..NSg-1.
__global__ __launch_bounds__(128)
void agg_full(const float* __restrict__ hs, const float* __restrict__ as_,
              const float* __restrict__ ad, const float* __restrict__ ea,
              const float* __restrict__ cbuf, const float* __restrict__ bias,
              float* __restrict__ acc, int init, int NSg, int NDg, int eS, int eD)
{
    __shared__ float a_sh[128];
    __shared__ float ex_sh[128];
    const int id = blockIdx.x;
    const int g  = id / NDg;
    const int j  = id - g * NDg;
    const int t  = threadIdx.x;
    const float c = cbuf[0];
    const int ebase = g * NSg * NDg + j * eD;
    const int sbase = g * NSg;
    const float adv = ad[id];
    if (t < NSg)
        a_sh[t] = lrelu2(as_[sbase + t] + adv + c * ea[ebase + t * eS]);
    __syncthreads();
    float amax = -INFINITY;
    for (int i = 0; i < NSg; ++i) amax = fmaxf(amax, a_sh[i]);
    if (t < NSg) ex_sh[t] = expf(a_sh[t] - amax);
    __syncthreads();
    float denom = 0.f, num = 0.f;
    for (int i = 0; i < NSg; ++i) {
        float e = ex_sh[i];
        denom += e;
        num += e * hs[(size_t)(sbase + i) * DD + t];
    }
    float val = num / (denom + 1e-16f) + bias[t];
    size_t o = (size_t)id * DD + t;
    if (init) acc[o] = val; else acc[o] += val;
}

// 1:1 mapping relation (vp_map / pv_map): dst j has one in-edge iff j < Vm.
__global__ __launch_bounds__(128)
void agg_map(const float* __restrict__ hs, const float* __restrict__ as_,
             const float* __restrict__ ad, const float* __restrict__ ea,
             const float* __restrict__ cbuf, const float* __restrict__ bias,
             float* __restrict__ acc, int init, int NSg, int NDg, int Vm)
{
    const int id = blockIdx.x;
    const int g  = id / NDg;
    const int j  = id - g * NDg;
    const int t  = threadIdx.x;
    float val = bias[t];
    if (j < Vm) {
        const int src = g * NSg + j;
        const int e   = g * Vm + j;
        float a = lrelu2(as_[src] + ad[id] + cbuf[0] * ea[e]);
        float ex = expf(a - a);              // single-edge softmax -> 1
        float w = ex / (ex + 1e-16f);
        val += w * hs[(size_t)src * DD + t];
    }
    size_t o = (size_t)id * DD + t;
    if (init) acc[o] = val; else acc[o] += val;
}

// vv ring (+/-1) with mean-attr self loop. V=10 per graph.
__global__ __launch_bounds__(128)
void agg_vv_k(const float* __restrict__ h, const float* __restrict__ as_,
              const float* __restrict__ ad, const float* __restrict__ ea,
              const float* __restrict__ cbuf, const float* __restrict__ bias,
              float* __restrict__ acc, int init)
{
    const int id = blockIdx.x;
    const int g = id / 10, v = id - g * 10;
    const int t = threadIdx.x;
    const int base = g * 10;
    const int s0 = base + (v + 9) % 10;
    const int s1 = base + (v + 1) % 10;
    const int e0 = base + (v + 9) % 10;        // block0: (v-1) -> v
    const int e1 = 2560 + base + v;            // block1: (v+1) -> v
    const float c = cbuf[0];
    const float adv = ad[id];
    const float ea0 = ea[e0], ea1 = ea[e1];
    const float eas = 0.5f * (ea0 + ea1);
    float a0 = lrelu2(as_[s0] + adv + c * ea0);
    float a1 = lrelu2(as_[s1] + adv + c * ea1);
    float a2 = lrelu2(as_[id] + adv + c * eas);
    float amax = fmaxf(a0, fmaxf(a1, a2));
    float x0 = expf(a0 - amax), x1 = expf(a1 - amax), x2 = expf(a2 - amax);
    float denom = x0 + x1 + x2 + 1e-16f;
    float num = x0 * h[(size_t)s0 * DD + t] + x1 * h[(size_t)s1 * DD + t]
              + x2 * h[(size_t)id * DD + t];
    float val = num / denom + bias[t];
    size_t o = (size_t)id * DD + t;
    if (init) acc[o] = val; else acc[o] += val;
}

// pp ring (+/-1, +/-5) with mean-attr self loop. P=100 per graph.
__global__ __launch_bounds__(128)
void agg_pp_k(const float* __restrict__ h, const float* __restrict__ as_,
              const float* __restrict__ ad, const float* __restrict__ ea,
              const float* __restrict__ cbuf, const float* __restrict__ bias,
              float* __restrict__ acc, int init)
{
    const int id = blockIdx.x;
    const int g = id / 100, p = id - g * 100;
    const int t = threadIdx.x;
    const int base = g * 100;
    const int sm1 = (p + 99) % 100, sp1 = (p + 1) % 100;
    const int sm5 = (p + 95) % 100, sp5 = (p + 5) % 100;
    const int e0 = base + sm1;                 // block0: (p-1) -> p
    const int e1 = 25600 + base + p;           // block1: (p+1) -> p
    const int e2 = 51200 + base + sm5;         // block2: (p-5) -> p
    const int e3 = 76800 + base + p;           // block3: (p+5) -> p
    const float c = cbuf[0];
    const float adv = ad[id];
    const float ea0 = ea[e0], ea1 = ea[e1], ea2 = ea[e2], ea3 = ea[e3];
    const float eas = 0.25f * (ea0 + ea1 + ea2 + ea3);
    float a0 = lrelu2(as_[base + sm1] + adv + c * ea0);
    float a1 = lrelu2(as_[base + sp1] + adv + c * ea1);
    float a2 = lrelu2(as_[base + sm5] + adv + c * ea2);
    float a3 = lrelu2(as_[base + sp5] + adv + c * ea3);
    float a4 = lrelu2(as_[id] + adv + c * eas);
    float amax = fmaxf(fmaxf(fmaxf(a0, a1), fmaxf(a2, a3)), a4);
    float x0 = expf(a0 - amax), x1 = expf(a1 - amax), x2 = expf(a2 - amax);
    float x3 = expf(a3 - amax), x4 = expf(a4 - amax);
    float denom = x0 + x1 + x2 + x3 + x4 + 1e-16f;
    float num = x0 * h[(size_t)(base + sm1) * DD + t]
              + x1 * h[(size_t)(base + sp1) * DD + t]
              + x2 * h[(size_t)(base + sm5) * DD + t]
              + x3 * h[(size_t)(base + sp5) * DD + t]
              + x4 * h[(size_t)id * DD + t];
    float val = num / denom + bias[t];
    size_t o = (size_t)id * DD + t;
    if (init) acc[o] = val; else acc[o] += val;
}

// out[n,d] = sum_k x[n,k]*W[d,k] + b[d]; optional second copy (p_init).
__global__ void init_proj(const float* __restrict__ x, const float* __restrict__ W,
                          const float* __restrict__ b, float* __restrict__ out,
                          float* __restrict__ out2, int N, int K)
{
    int i = blockIdx.x * 256 + threadIdx.x;
    if (i >= N * DD) return;
    int n = i >> 7, d = i & 127;
    float s = b[d];
    for (int k = 0; k < K; ++k) s += x[n * K + k] * W[d * K + k];
    out[i] = s;
    if (out2) out2[i] = s;
}

__global__ void lrelu01_k(const float* __restrict__ in, float* __restrict__ out, int n)
{
    int i = blockIdx.x * 256 + threadIdx.x;
    if (i < n) { float v = in[i]; out[i] = v > 0.f ? v : 0.01f * v; }
}

extern "C" void kernel_launch(void* const* d_in, const int* in_sizes, int n_in,
                              void* d_out, int out_size, void* d_ws, size_t ws_size,
                              hipStream_t stream)
{
    (void)in_sizes; (void)n_in; (void)out_size; (void)ws_size;
    auto F = [&](int i) { return (const float*)d_in[i]; };

    const float* x_v = F(0);
    const float* x_p = F(1);
    const float* ea_vp_im  = F(3);
    const float* ea_pv_im  = F(5);
    const float* ea_vp_map = F(7);
    const float* ea_pv_map = F(9);
    const float* ea_vv     = F(11);
    const float* ea_pp     = F(13);

    const int NP = 25600, NV = 2560;
    float* ws = (float*)d_ws;
    float* hp      = ws;
    float* hv      = hp + (size_t)NP * DD;
    float* p_init  = hv + (size_t)NV * DD;
    float* acc_p   = p_init + (size_t)NP * DD;
    float* acc_v   = acc_p + (size_t)NP * DD;
    float* projS   = acc_v + (size_t)NV * DD;
    float* projD   = projS + (size_t)NP * DD;
    float* alpha_s = projD + (size_t)NP * DD;
    float* alpha_d = alpha_s + NP;
    float* cbuf    = alpha_d + NP;

    // initial projections (params: 14=init_p.W 15=init_p.b 16=init_v.W 17=init_v.b)
    init_proj<<<(NP * DD + 255) / 256, 256, 0, stream>>>(x_p, F(14), F(15), hp, p_init, NP, 9);
    init_proj<<<(NV * DD + 255) / 256, 256, 0, stream>>>(x_v, F(16), F(17), hv, nullptr, NV, 7);

    auto gemm = [&](const float* A, const float* W, float* C, int M,
                    const float* bias, const float* res) {
        gemm_nt_wmma<<<dim3(M / 16, 2), 32, 0, stream>>>(A, W, C, M, bias, res);
    };
    auto dots = [&](const float* H, const float* att, float* o, int N) {
        rowdot<<<N / 4, 128, 0, stream>>>(H, att, o, N);
    };

    for (int l = 0; l < 3; ++l) {
        const int L = 18 + 40 * l;  // layer leaves: pp(6) pv_im(7) pv_map(7) vp_im(7) vp_map(7) vv(6)

        // ---- P side: new_p = GAT_vp_im + GAT_vp_map + GAT_pp ----
        {   // vp_im: Wd=+0 We=+1 Ws=+2 att_d=+3 att_e=+4 att_s=+5 b=+6
            int b0 = L + 20;
            gemm(hv, F(b0 + 2), projS, NV, nullptr, nullptr);
            gemm(hp, F(b0 + 0), projD, NP, nullptr, nullptr);
            dots(projS, F(b0 + 5), alpha_s, NV);
            dots(projD, F(b0 + 3), alpha_d, NP);
            dotc_kernel<<<1, 32, 0, stream>>>(F(b0 + 4), F(b0 + 1), cbuf);
            agg_full<<<NP, 128, 0, stream>>>(projS, alpha_s, alpha_d, ea_vp_im, cbuf,
                                             F(b0 + 6), acc_p, 1, 10, 100, 100, 1);
        }
        {   // vp_map
            int b0 = L + 27;
            gemm(hv, F(b0 + 2), projS, NV, nullptr, nullptr);
            gemm(hp, F(b0 + 0), projD, NP, nullptr, nullptr);
            dots(projS, F(b0 + 5), alpha_s, NV);
            dots(projD, F(b0 + 3), alpha_d, NP);
            dotc_kernel<<<1, 32, 0, stream>>>(F(b0 + 4), F(b0 + 1), cbuf);
            agg_map<<<NP, 128, 0, stream>>>(projS, alpha_s, alpha_d, ea_vp_map, cbuf,
                                            F(b0 + 6), acc_p, 0, 10, 100, 10);
        }
        {   // pp (shared): W=+0 We=+1 att_d=+2 att_e=+3 att_s=+4 b=+5
            int b0 = L;
            gemm(hp, F(b0 + 0), projS, NP, nullptr, nullptr);
            dots(projS, F(b0 + 4), alpha_s, NP);
            dots(projS, F(b0 + 2), alpha_d, NP);
            dotc_kernel<<<1, 32, 0, stream>>>(F(b0 + 3), F(b0 + 1), cbuf);
            agg_pp_k<<<NP, 128, 0, stream>>>(projS, alpha_s, alpha_d, ea_pp, cbuf,
                                             F(b0 + 5), acc_p, 0);
        }

        // ---- V side (output-independent for last layer) ----
        if (l < 2) {
            {   // pv_im
                int b0 = L + 6;
                gemm(hp, F(b0 + 2), projS, NP, nullptr, nullptr);
                gemm(hv, F(b0 + 0), projD, NV, nullptr, nullptr);
                dots(projS, F(b0 + 5), alpha_s, NP);
                dots(projD, F(b0 + 3), alpha_d, NV);
                dotc_kernel<<<1, 32, 0, stream>>>(F(b0 + 4), F(b0 + 1), cbuf);
                agg_full<<<NV, 128, 0, stream>>>(projS, alpha_s, alpha_d, ea_pv_im, cbuf,
                                                 F(b0 + 6), acc_v, 1, 100, 10, 1, 100);
            }
            {   // pv_map
                int b0 = L + 13;
                gemm(hp, F(b0 + 2), projS, NP, nullptr, nullptr);
                gemm(hv, F(b0 + 0), projD, NV, nullptr, nullptr);
                dots(projS, F(b0 + 5), alpha_s, NP);
                dots(projD, F(b0 + 3), alpha_d, NV);
                dotc_kernel<<<1, 32, 0, stream>>>(F(b0 + 4), F(b0 + 1), cbuf);
                agg_map<<<NV, 128, 0, stream>>>(projS, alpha_s, alpha_d, ea_pv_map, cbuf,
                                                F(b0 + 6), acc_v, 0, 100, 10, 10);
            }
            {   // vv (shared)
                int b0 = L + 34;
                gemm(hv, F(b0 + 0), projS, NV, nullptr, nullptr);
                dots(projS, F(b0 + 4), alpha_s, NV);
                dots(projS, F(b0 + 2), alpha_d, NV);
                dotc_kernel<<<1, 32, 0, stream>>>(F(b0 + 3), F(b0 + 1), cbuf);
                agg_vv_k<<<NV, 128, 0, stream>>>(projS, alpha_s, alpha_d, ea_vv, cbuf,
                                                 F(b0 + 5), acc_v, 0);
            }
        }

        lrelu01_k<<<(NP * DD + 255) / 256, 256, 0, stream>>>(acc_p, hp, NP * DD);
        if (l < 2)
            lrelu01_k<<<(NV * DD + 255) / 256, 256, 0, stream>>>(acc_v, hv, NV * DD);
    }

    // p_out = hp @ out_W^T + out_b + p_init  (params: 138=out_p.W, 139=out_p.b)
    gemm(hp, F(138), (float*)d_out, NP, F(139), p_init);
}